// VectorQuantizerEMA_43731357008523
// MI455X (gfx1250) — compile-verified
//
#include <hip/hip_runtime.h>

typedef __attribute__((ext_vector_type(2))) float v2f;
typedef __attribute__((ext_vector_type(8))) float v8f;

#define KCODES 256
#define DDIM 128
#define TPB 1024            // tokens per batch (H*W)
#define NTOK 131072         // total tokens
#define NELEM 16777216      // total z elements

// workspace layout (float offsets)
#define WS_CNORM    0       // 256
#define WS_ENCSUM   256     // 256
#define WS_SQSUM    512     // 1
#define WS_ENCBATCH 768     // 32768
#define WS_FLOATS   33536

// -------- kernel 0: codebook squared norms --------
__global__ __launch_bounds__(256)
void vq_cnorm_kernel(const float* __restrict__ cb, float* __restrict__ cnorm) {
    const int k = threadIdx.x;
    const float* row = cb + k * DDIM;
    float s = 0.f;
#pragma unroll 8
    for (int j = 0; j < DDIM; ++j) { float v = row[j]; s += v * v; }
    cnorm[k] = s;
}

// -------- kernel 1: fp32 WMMA distance + argmin --------
// one wave handles 16 tokens vs all 256 codes; scores = codebook @ z_slab.
// The wave's entire z tile (16 tok x 128 dim) is held in registers as the 32
// pre-formatted WMMA B-operands; each code tile then runs 32 back-to-back
// v_wmma_f32_16x16x4_f32 fed only by contiguous codebook b64 loads.
__global__ __launch_bounds__(128)
void vq_argmin_kernel(const float* __restrict__ z, const float* __restrict__ cb,
                      const float* __restrict__ cnorm, float* __restrict__ out_idx) {
    const int lane = threadIdx.x & 31;
    const int wid  = threadIdx.x >> 5;
    const int b    = blockIdx.y;
    const int t0   = (blockIdx.x * 4 + wid) * 16;
    const int half = lane >> 4;                 // 0: lanes 0-15, 1: lanes 16-31
    const int l16  = lane & 15;
    const float* zb = z + (size_t)b * (DDIM * TPB);

    // Preload B operands: bz[j] covers k = 4j .. 4j+3 (this lane holds 2 of 4)
    const float* bcol = zb + (size_t)(half << 1) * TPB + t0 + l16;
    v2f bz[32];
#pragma unroll
    for (int j = 0; j < 32; ++j) {
        bz[j] = (v2f){ bcol[(size_t)(4 * j) * TPB],
                       bcol[(size_t)(4 * j + 1) * TPB] };
    }

    float minval = 3.402823466e38f;
    int   minidx = 0;

    for (int ct = 0; ct < 16; ++ct) {           // 16 code tiles of 16
        v8f acc = {};
        // A tile: codebook rows (16 codes x 4 dims per step), contiguous pairs
        const float* arow = cb + (size_t)(ct * 16 + l16) * DDIM + (half << 1);
#pragma unroll
        for (int j = 0; j < 32; ++j) {
            v2f a = (v2f){ arow[4 * j], arow[4 * j + 1] };
            acc = __builtin_amdgcn_wmma_f32_16x16x4_f32(
                false, a, false, bz[j], (short)0, acc, false, false);
        }
        // D layout: lanes<16 -> codes ct*16+v, lanes>=16 -> codes ct*16+8+v
        const int mbase = ct * 16 + (half << 3);
#pragma unroll
        for (int v = 0; v < 8; ++v) {
            const int m = mbase + v;
            const float d = cnorm[m] - 2.0f * acc[v];   // |z|^2 constant per token
            if (d < minval) { minval = d; minidx = m; }
        }
    }
    // combine the two half-wave candidates for each token (argmin tie -> lowest idx)
    const float ov = __shfl_xor(minval, 16, 32);
    const int   oi = __shfl_xor(minidx, 16, 32);
    if (ov < minval || (ov == minval && oi < minidx)) { minval = ov; minidx = oi; }
    if (lane < 16) out_idx[b * TPB + t0 + l16] = (float)minidx;  // exact for k<256
}

// -------- kernel 2: gather, straight-through output, MSE, segment sums --------
__global__ __launch_bounds__(256)
void vq_gather_kernel(const float* __restrict__ z, const float* __restrict__ cb,
                      const float* __restrict__ idxf, float* __restrict__ out_qst,
                      float* __restrict__ enc_sum, float* __restrict__ enc_batch,
                      float* __restrict__ sqsum) {
    __shared__ float red[256];
    const int tid = threadIdx.x;
    const unsigned gid = blockIdx.x * 256u + tid;
    const int d = gid >> 17;                 // feature dim (slow)
    const int n = gid & (NTOK - 1);          // token (fast -> coalesced z)
    const int b = n >> 10;
    const int t = n & 1023;
    const size_t za = (size_t)b * (DDIM * TPB) + (size_t)d * TPB + t;
    const int   k  = (int)idxf[n];
    const float zv = z[za];
    const float q  = cb[k * DDIM + d];
    out_qst[za] = zv + (q - zv);             // straight-through forward
    const float e = zv - q;
    atomicAdd(&enc_batch[k * DDIM + d], zv);
    if (d == 0) atomicAdd(&enc_sum[k], 1.0f);
    red[tid] = e * e;
    __syncthreads();
    for (int s = 128; s > 0; s >>= 1) {
        if (tid < s) red[tid] += red[tid + s];
        __syncthreads();
    }
    if (tid == 0) atomicAdd(sqsum, red[0]);
}

// -------- kernel 3: EMA finalize + losses (single block) --------
__global__ __launch_bounds__(256)
void vq_final_kernel(const float* __restrict__ ema_count, const float* __restrict__ ema_weight,
                     const float* __restrict__ enc_sum, const float* __restrict__ enc_batch,
                     const float* __restrict__ sqsum,
                     float* __restrict__ o_commit, float* __restrict__ o_cbloss,
                     float* __restrict__ o_newcb, float* __restrict__ o_newcount,
                     float* __restrict__ o_newweight) {
    __shared__ float red[256];
    const int k = threadIdx.x;
    const float nc = 0.99f * ema_count[k] + 0.01f * enc_sum[k];
    o_newcount[k] = nc;
    red[k] = nc;
    __syncthreads();
    for (int s = 128; s > 0; s >>= 1) { if (k < s) red[k] += red[k + s]; __syncthreads(); }
    const float ntot  = red[0];
    const float normc = (nc + 1e-5f) / (ntot + (float)KCODES * 1e-5f) * ntot;
#pragma unroll 4
    for (int j = 0; j < DDIM; ++j) {
        const float w = 0.99f * ema_weight[k * DDIM + j] + 0.01f * enc_batch[k * DDIM + j];
        o_newweight[k * DDIM + j] = w;
        o_newcb[k * DDIM + j]     = w / normc;
    }
    if (k == 0) {
        const float mse = sqsum[0] / (float)NELEM;
        o_cbloss[0] = mse;
        o_commit[0] = 0.25f * mse;
    }
}

extern "C" void kernel_launch(void* const* d_in, const int* in_sizes, int n_in,
                              void* d_out, int out_size, void* d_ws, size_t ws_size,
                              hipStream_t stream) {
    const float* z          = (const float*)d_in[0];
    const float* cb         = (const float*)d_in[1];
    const float* ema_count  = (const float*)d_in[2];
    const float* ema_weight = (const float*)d_in[3];
    float* out = (float*)d_out;
    float* ws  = (float*)d_ws;

    // output slots (return-order flat concat)
    float* o_qst       = out;                       // 16777216
    float* o_idx       = out + 16777216;            // 131072 (float-encoded ints)
    float* o_commit    = out + 16908288;            // 1
    float* o_cbloss    = out + 16908289;            // 1
    float* o_newcb     = out + 16908290;            // 32768
    float* o_newcount  = out + 16941058;            // 256
    float* o_newweight = out + 16941314;            // 32768

    hipMemsetAsync(ws, 0, WS_FLOATS * sizeof(float), stream);
    vq_cnorm_kernel<<<1, 256, 0, stream>>>(cb, ws + WS_CNORM);
    vq_argmin_kernel<<<dim3(16, 128), 128, 0, stream>>>(z, cb, ws + WS_CNORM, o_idx);
    vq_gather_kernel<<<NELEM / 256, 256, 0, stream>>>(z, cb, o_idx, o_qst,
                                                      ws + WS_ENCSUM, ws + WS_ENCBATCH,
                                                      ws + WS_SQSUM);
    vq_final_kernel<<<1, 256, 0, stream>>>(ema_count, ema_weight,
                                           ws + WS_ENCSUM, ws + WS_ENCBATCH, ws + WS_SQSUM,
                                           o_commit, o_cbloss, o_newcb, o_newcount, o_newweight);
}